// SinkhornNet_34359738798
// MI455X (gfx1250) — compile-verified
//
#include <hip/hip_runtime.h>

// ---------------------------------------------------------------------------
// SinkhornNet for MI455X (gfx1250, wave32)
//   B=131072, K=6, S=5, D_LAT=16, D_FEAT=8, N_ITERS=5
// Memory-bound (~257MB HBM traffic @23.3TB/s ~ 11us); Sinkhorn done in linear
// domain (mathematically identical to log-domain logsumexp normalization) to
// minimize transcendental pressure. The sinknet Linear is a real GEMM and is
// done with V_WMMA_F32_16X16X4_F32.
// ---------------------------------------------------------------------------

namespace {
constexpr int KKc    = 6;
constexpr int KK2c   = 36;
constexpr int DLATc  = 16;
constexpr int NITERc = 5;
}

typedef __attribute__((ext_vector_type(2))) float v2f;
typedef __attribute__((ext_vector_type(8))) float v8f;

// ---------------------------------------------------------------------------
// Kernel 1: la_ws[B,36] = latent[B,16] @ W_sink[16,36] + b_sink  (WMMA f32)
// One wave computes a 16-row x 48-col tile (3 N-tiles of 16, cols 36..47 are
// padding and never stored). K=16 is consumed as 4 WMMA K-steps of 4.
// Bias is pre-seeded into the C accumulator (all 8 D elements of a lane share
// one column). Full tiles (the common case) take an unconditional store path.
// ---------------------------------------------------------------------------
__global__ __launch_bounds__(256) void sinknet_wmma_kernel(
    const float* __restrict__ latent, const float* __restrict__ Ws,
    const float* __restrict__ bs, float* __restrict__ la_ws, int Btot)
{
  const int lane = threadIdx.x & 31;
  const int wave = threadIdx.x >> 5;
  const int tile = blockIdx.x * 8 + wave;     // 16-row tile index
  const int row0 = tile * 16;
  if (row0 >= Btot) return;                   // wave-uniform: EXEC stays all-1s
  const bool fullTile = (row0 + 16 <= Btot);  // wave-uniform

  const int r  = lane & 15;                   // A: row-in-tile, B/D: col-in-tile
  const int hi = lane >> 4;                   // lane group 0/1
  const int kb = hi << 1;                     // K sub-offset {0,2}

  // A fragments: lane holds latent[row0+r, ks*4+kb +{0,1}] for 4 K-steps
  int arow = row0 + r; if (arow >= Btot) arow = Btot - 1;
  v2f a[4];
  #pragma unroll
  for (int ks = 0; ks < 4; ++ks) {
    a[ks] = *(const v2f*)(latent + (size_t)arow * DLATc + ks * 4 + kb);
  }

  #pragma unroll
  for (int nt = 0; nt < 3; ++nt) {
    const int n0  = nt * 16;
    const int col = n0 + r;
    const bool cok = (nt < 2) || (col < KK2c);  // nt<2 -> col<=31<36 always

    // Seed accumulator with bias (every D element of this lane is column `col`)
    const float bv = cok ? bs[col] : 0.f;
    v8f acc = {bv, bv, bv, bv, bv, bv, bv, bv};

    #pragma unroll
    for (int ks = 0; ks < 4; ++ks) {
      const int k = ks * 4 + kb;
      v2f bf;
      bf.x = cok ? Ws[(size_t)k * KK2c + col]       : 0.f;
      bf.y = cok ? Ws[(size_t)(k + 1) * KK2c + col] : 0.f;
      acc = __builtin_amdgcn_wmma_f32_16x16x4_f32(
          false, a[ks], false, bf, (short)0, acc, false, false);
    }

    if (fullTile) {
      #pragma unroll
      for (int v = 0; v < 8; ++v) {
        const int row = row0 + v + (hi << 3); // D layout: lanes16-31 -> M=v+8
        if (cok) la_ws[(size_t)row * KK2c + col] = acc[v];
      }
    } else {
      #pragma unroll
      for (int v = 0; v < 8; ++v) {
        const int row = row0 + v + (hi << 3);
        if (cok && row < Btot) la_ws[(size_t)row * KK2c + col] = acc[v];
      }
    }
  }
}

// ---------------------------------------------------------------------------
// Kernel 2: per (sample, batch) matrix: gumbel + Sinkhorn(5) + P^T @ seq.
// Linear-domain Sinkhorn: exp once, then alternating row/col sum-normalize.
// grid = (ceil(B/256), NSAMPLES); b = x index, s = blockIdx.y.
// la_ws may be null (ws too small) -> recompute log_alpha with FMAs.
// ---------------------------------------------------------------------------
__global__ __launch_bounds__(256) void sinkhorn_kernel(
    const float* __restrict__ la_ws, const float* __restrict__ latent,
    const float* __restrict__ Ws, const float* __restrict__ bs,
    const float* __restrict__ noise, const float* __restrict__ seq,
    float* __restrict__ ordered, int Btot)
{
  const int b = blockIdx.x * blockDim.x + threadIdx.x;
  if (b >= Btot) return;
  const size_t tid = (size_t)blockIdx.y * (size_t)Btot + (size_t)b;

  // --- log_alpha row (36 floats) ---
  float lw[KK2c];
  if (la_ws) {
    const float4* lp = (const float4*)(la_ws + (size_t)b * KK2c);
    #pragma unroll
    for (int q = 0; q < 9; ++q) {
      float4 v = lp[q];
      lw[q * 4 + 0] = v.x; lw[q * 4 + 1] = v.y;
      lw[q * 4 + 2] = v.z; lw[q * 4 + 3] = v.w;
    }
  } else {
    float x[DLATc];
    const float4* xp = (const float4*)(latent + (size_t)b * DLATc);
    #pragma unroll
    for (int q = 0; q < 4; ++q) {
      float4 v = xp[q];
      x[q * 4 + 0] = v.x; x[q * 4 + 1] = v.y;
      x[q * 4 + 2] = v.z; x[q * 4 + 3] = v.w;
    }
    #pragma unroll
    for (int e = 0; e < KK2c; ++e) {
      float t = bs[e];
      #pragma unroll
      for (int i = 0; i < DLATc; ++i) t = fmaf(x[i], Ws[i * KK2c + e], t);
      lw[e] = t;
    }
  }

  // --- gumbel perturb + exp (linear domain) ---
  float A[KK2c];
  const float4* np = (const float4*)(noise + tid * KK2c);
  #pragma unroll
  for (int q = 0; q < 9; ++q) {
    float4 v = np[q];
    float u[4] = {v.x, v.y, v.z, v.w};
    #pragma unroll
    for (int t = 0; t < 4; ++t) {
      // gumbel = -log(EPS - log(u + EPS)); la = log_alpha + gumbel (NF=T=1)
      const float g = -__logf(1e-20f - __logf(u[t] + 1e-20f));
      A[q * 4 + t] = __expf(lw[q * 4 + t] + g);
    }
  }

  // --- Sinkhorn iterations: row-normalize (axis=2) then col-normalize ---
  #pragma unroll
  for (int it = 0; it < NITERc; ++it) {
    #pragma unroll
    for (int i = 0; i < KKc; ++i) {
      const float s = A[i*6+0] + A[i*6+1] + A[i*6+2] + A[i*6+3] + A[i*6+4] + A[i*6+5];
      const float inv = __builtin_amdgcn_rcpf(s);
      #pragma unroll
      for (int j = 0; j < KKc; ++j) A[i*6+j] *= inv;
    }
    #pragma unroll
    for (int j = 0; j < KKc; ++j) {
      const float s = A[0*6+j] + A[1*6+j] + A[2*6+j] + A[3*6+j] + A[4*6+j] + A[5*6+j];
      const float inv = __builtin_amdgcn_rcpf(s);
      #pragma unroll
      for (int i = 0; i < KKc; ++i) A[i*6+j] *= inv;
    }
  }

  // --- ordered[i][j] = sum_k sink[k][i] * seq[k][j]  (P = sink^T) ---
  float S[48];
  const float4* sp = (const float4*)(seq + (size_t)b * 48);
  #pragma unroll
  for (int q = 0; q < 12; ++q) {
    float4 v = sp[q];
    S[q * 4 + 0] = v.x; S[q * 4 + 1] = v.y;
    S[q * 4 + 2] = v.z; S[q * 4 + 3] = v.w;
  }
  float4* op = (float4*)(ordered + tid * 48);
  #pragma unroll
  for (int i = 0; i < KKc; ++i) {
    float acc[8] = {0, 0, 0, 0, 0, 0, 0, 0};
    #pragma unroll
    for (int k = 0; k < KKc; ++k) {
      const float p = A[k * 6 + i];
      #pragma unroll
      for (int j = 0; j < 8; ++j) acc[j] = fmaf(p, S[k * 8 + j], acc[j]);
    }
    op[i * 2 + 0] = make_float4(acc[0], acc[1], acc[2], acc[3]);
    op[i * 2 + 1] = make_float4(acc[4], acc[5], acc[6], acc[7]);
  }
}

// ---------------------------------------------------------------------------
// Kernel 3: stopping[b,:] = softmax(latent[b] @ W_mask + b_mask)
// ---------------------------------------------------------------------------
__global__ __launch_bounds__(256) void masknet_kernel(
    const float* __restrict__ latent, const float* __restrict__ Wm,
    const float* __restrict__ bm, float* __restrict__ stopping, int Btot)
{
  const int b = blockIdx.x * blockDim.x + threadIdx.x;
  if (b >= Btot) return;
  float x[DLATc];
  const float4* xp = (const float4*)(latent + (size_t)b * DLATc);
  #pragma unroll
  for (int q = 0; q < 4; ++q) {
    float4 v = xp[q];
    x[q * 4 + 0] = v.x; x[q * 4 + 1] = v.y;
    x[q * 4 + 2] = v.z; x[q * 4 + 3] = v.w;
  }
  float o[KKc];
  #pragma unroll
  for (int j = 0; j < KKc; ++j) o[j] = bm[j];
  #pragma unroll
  for (int i = 0; i < DLATc; ++i) {
    #pragma unroll
    for (int j = 0; j < KKc; ++j) o[j] = fmaf(x[i], Wm[i * KKc + j], o[j]);
  }
  float m = o[0];
  #pragma unroll
  for (int j = 1; j < KKc; ++j) m = fmaxf(m, o[j]);
  float s = 0.f;
  #pragma unroll
  for (int j = 0; j < KKc; ++j) { o[j] = __expf(o[j] - m); s += o[j]; }
  const float inv = __builtin_amdgcn_rcpf(s);
  float* op = stopping + (size_t)b * KKc;
  #pragma unroll
  for (int j = 0; j < KKc; ++j) op[j] = o[j] * inv;
}

// ---------------------------------------------------------------------------
extern "C" void kernel_launch(void* const* d_in, const int* in_sizes, int n_in,
                              void* d_out, int out_size, void* d_ws, size_t ws_size,
                              hipStream_t stream) {
  const float* latent = (const float*)d_in[0];
  const float* seq    = (const float*)d_in[1];
  const float* noise  = (const float*)d_in[2];
  const float* Wsink  = (const float*)d_in[3];
  const float* bsink  = (const float*)d_in[4];
  const float* Wmask  = (const float*)d_in[5];
  const float* bmask  = (const float*)d_in[6];

  const int Btot  = in_sizes[0] / DLATc;       // 131072
  const int total = in_sizes[2] / KK2c;        // 5*B
  const int ns    = total / Btot;              // 5

  float* ordered  = (float*)d_out;
  float* stopping = (float*)d_out + (size_t)total * 48;

  const size_t ws_need = (size_t)Btot * KK2c * sizeof(float);
  float* la_ws = (ws_size >= ws_need) ? (float*)d_ws : nullptr;

  if (la_ws) {
    const int tiles  = (Btot + 15) / 16;
    const int blocks = (tiles + 7) / 8;        // 8 waves (tiles) per block
    sinknet_wmma_kernel<<<blocks, 256, 0, stream>>>(latent, Wsink, bsink, la_ws, Btot);
  }

  {
    dim3 grid((Btot + 255) / 256, ns);
    sinkhorn_kernel<<<grid, 256, 0, stream>>>(la_ws, latent, Wsink, bsink,
                                              noise, seq, ordered, Btot);
  }
  {
    const int blocks = (Btot + 255) / 256;
    masknet_kernel<<<blocks, 256, 0, stream>>>(latent, Wmask, bmask, stopping, Btot);
  }
}